// SAGESparseLayer_54863912239193
// MI455X (gfx1250) — compile-verified
//
#include <hip/hip_runtime.h>
#include <hip/hip_bf16.h>

typedef __attribute__((ext_vector_type(2))) float v2f;
typedef __attribute__((ext_vector_type(8))) float v8f;

#define N_NODES 50000
#define N_EDGES 800000
#define D_IN    128
#define D_OUT   128
#define D_K     256     // 2 * D_IN
#define HSTRIDE 260     // LDS row stride (floats): 260 % 64 == 4 -> conflict-free, 16B-aligned rows

// ---------------------------------------------------------------------------
// Kernel 1: zero the segment-sum accumulator + counts in workspace.
// ---------------------------------------------------------------------------
__global__ void sage_zero_ws_kernel(float* __restrict__ sums, float* __restrict__ cnts) {
    const long long stride = (long long)gridDim.x * blockDim.x;
    long long i = (long long)blockIdx.x * blockDim.x + threadIdx.x;
    const long long total = (long long)N_NODES * D_IN;
    for (long long j = i; j < total; j += stride) sums[j] = 0.0f;
    for (long long j = i; j < N_NODES; j += stride) cnts[j] = 0.0f;
}

// ---------------------------------------------------------------------------
// Kernel 2: edge scatter. One wave per edge; lane l owns float4 [4l..4l+3]
// of the 128-float feature row -> fully coalesced 512B gather per edge and
// contiguous-128B groups of f32 atomics into the L2-resident accumulator.
// ---------------------------------------------------------------------------
__global__ __launch_bounds__(256)
void sage_edge_scatter_kernel(const float*     __restrict__ feat,
                              const long long* __restrict__ idx,   // [2, E]: row0=dst, row1=src
                              const float*     __restrict__ ew,    // [E]
                              float*           __restrict__ sums,  // [N, 128]
                              float*           __restrict__ cnts)  // [N]
{
    const int lane  = threadIdx.x & 31;
    const int wpb   = blockDim.x >> 5;
    const int wave  = blockIdx.x * wpb + (threadIdx.x >> 5);
    const int nwave = gridDim.x * wpb;

    for (int e = wave; e < N_EDGES; e += nwave) {
        const long long dst = idx[e];
        const long long src = idx[(long long)N_EDGES + e];
        const float     wt  = ew[e];

        const float4 v = ((const float4*)(feat + src * (long long)D_IN))[lane];

        float* d = sums + dst * (long long)D_IN + lane * 4;
        atomicAdd(d + 0, v.x * wt);
        atomicAdd(d + 1, v.y * wt);
        atomicAdd(d + 2, v.z * wt);
        atomicAdd(d + 3, v.w * wt);
        if (lane == 0) atomicAdd(cnts + dst, 1.0f);
    }
}

// ---------------------------------------------------------------------------
// Kernel 3: fused mean + concat + GEMM via V_WMMA_F32_16X16X4_F32.
// Block = 256 threads = 8 waves, handles 16 nodes. h (16x256 f32) staged in
// LDS; wave w computes output columns [16w, 16w+16) with a K-loop of 64 WMMAs.
// ---------------------------------------------------------------------------
__global__ __launch_bounds__(256)
void sage_gemm_kernel(const float* __restrict__ feat,   // [N, 128]
                      const float* __restrict__ sums,   // [N, 128]
                      const float* __restrict__ cnts,   // [N]
                      const float* __restrict__ Wm,     // [256, 128] row-major
                      const float* __restrict__ bias,   // [128]
                      float*       __restrict__ out)    // [N, 128]
{
    __shared__ float h[16 * HSTRIDE];

    const int rowBase = blockIdx.x * 16;
    const int tid     = threadIdx.x;

    // Stage h = [feature | sums/max(cnt,1)] : per half 16x128 f32 = 512 float4s.
    for (int q = tid; q < 512; q += 256) {
        const int r    = q >> 5;     // node row 0..15
        const int c4   = q & 31;     // float4 column 0..31
        const int node = rowBase + r;

        const float4 f = ((const float4*)(feat + (long long)node * D_IN))[c4];
        float* p0 = &h[r * HSTRIDE + c4 * 4];
        p0[0] = f.x; p0[1] = f.y; p0[2] = f.z; p0[3] = f.w;

        const float4 s   = ((const float4*)(sums + (long long)node * D_IN))[c4];
        const float  inv = 1.0f / fmaxf(cnts[node], 1.0f);
        float* p1 = &h[r * HSTRIDE + D_IN + c4 * 4];
        p1[0] = s.x * inv; p1[1] = s.y * inv; p1[2] = s.z * inv; p1[3] = s.w * inv;
    }
    __syncthreads();

    const int lane    = tid & 31;
    const int waveN   = tid >> 5;         // 0..7
    const int colBase = waveN * 16;
    const int mr      = lane & 15;        // M row (A) / N col (B,D) within tile
    const int halfK   = (lane >> 4) * 2;  // lanes 16-31 carry K+2, K+3

    v8f c = {};
    for (int k0 = 0; k0 < D_K; k0 += 4) {
        const int kk = k0 + halfK;

        // A fragment: A[mr][kk], A[mr][kk+1] from LDS (8B-aligned).
        const float2 af = *(const float2*)&h[mr * HSTRIDE + kk];
        v2f a; a[0] = af.x; a[1] = af.y;

        // B fragment: W[kk][colBase+mr], W[kk+1][colBase+mr] (L2-hot).
        v2f b;
        b[0] = Wm[kk * D_OUT + colBase + mr];
        b[1] = Wm[(kk + 1) * D_OUT + colBase + mr];

        // D = A * B + C  (16x16 f32 accumulator, 8 VGPRs)
        c = __builtin_amdgcn_wmma_f32_16x16x4_f32(
                /*neg_a=*/false, a, /*neg_b=*/false, b,
                /*c_mod=*/(short)0, c, /*reuse_a=*/false, /*reuse_b=*/false);
    }

    // D layout: VGPR v -> M = v + 8*(lane>=16), N = lane&15.
    const float bv    = bias[colBase + mr];
    const int   mbase = (lane >> 4) * 8;
#pragma unroll
    for (int v = 0; v < 8; ++v) {
        const int m = mbase + v;
        out[(long long)(rowBase + m) * D_OUT + colBase + mr] = c[v] + bv;
    }
}

// ---------------------------------------------------------------------------
extern "C" void kernel_launch(void* const* d_in, const int* in_sizes, int n_in,
                              void* d_out, int out_size, void* d_ws, size_t ws_size,
                              hipStream_t stream) {
    const float*     feat = (const float*)d_in[0];      // [50000,128] f32
    const long long* idx  = (const long long*)d_in[1];  // [2,800000] i64
    const float*     ew   = (const float*)d_in[2];      // [800000,1] f32
    const float*     Wm   = (const float*)d_in[3];      // [256,128]  f32
    const float*     bias = (const float*)d_in[4];      // [128]      f32
    float*           out  = (float*)d_out;              // [50000,128] f32

    float* sums = (float*)d_ws;                              // 25.6 MB
    float* cnts = sums + (size_t)N_NODES * D_IN;             // +0.2 MB

    sage_zero_ws_kernel<<<512, 256, 0, stream>>>(sums, cnts);
    sage_edge_scatter_kernel<<<2048, 256, 0, stream>>>(feat, idx, ew, sums, cnts);
    sage_gemm_kernel<<<N_NODES / 16, 256, 0, stream>>>(feat, sums, cnts, Wm, bias, out);
}